// GNN_5866925326812
// MI455X (gfx1250) — compile-verified
//
#include <hip/hip_runtime.h>
#include <cstdint>

// CDNA5 / gfx1250: wave32, WMMA 16x16x4 f32 (full-precision matrix pipe).
typedef float v2f __attribute__((ext_vector_type(2)));
typedef float v8f __attribute__((ext_vector_type(8)));

__device__ __forceinline__ v8f wmma_f32(v2f a, v2f b, v8f c) {
  // (neg_a, A, neg_b, B, c_mod, C, reuse_a, reuse_b)
  return __builtin_amdgcn_wmma_f32_16x16x4_f32(false, a, false, b, (short)0, c,
                                               false, false);
}

__device__ __forceinline__ float sigmoidf_(float x) {
  return 1.0f / (1.0f + __expf(-x));
}

// ---------------------------------------------------------------------------
// Edge kernel: fused 3-gate message MLP + ReLU + atomic scatter.
//   m_g = relu( [x[src], e] @ Wm_g(effective) + bm_g );  agg_g[dst] += m_g
// Exploits h_prev == 0: the 64 h-rows of Wm are skipped (effective K = D+4).
// B is stored K-pair interleaved: B[k][col] -> sB[(k>>1)*PSTR + 2*col + (k&1)]
// so each lane's WMMA B fragment is one 8B-aligned ds_load_b64 (no repack
// movs); PSTR % 64 == 32 dwords keeps the two 16-lane bank runs disjoint.
// ---------------------------------------------------------------------------
template <int D>
__launch_bounds__(256)
__global__ void edge_msg_kernel(const float* __restrict__ x,      // [N, D]
                                const int* __restrict__ src,
                                const int* __restrict__ dst,
                                const float* __restrict__ eattr,  // [E, 4]
                                const float* __restrict__ Wm0,
                                const float* __restrict__ bm0,
                                const float* __restrict__ Wm1,
                                const float* __restrict__ bm1,
                                const float* __restrict__ Wm2,
                                const float* __restrict__ bm2,
                                float* __restrict__ agg,          // [3][N*64]
                                int N, int E) {
  constexpr int K = D + 4;
  constexpr int KPAD = (K + 3) & ~3;
  constexpr int ASTR = (KPAD % 16 == 0) ? KPAD + 4 : KPAD;  // conflict-free
  constexpr int PSTR = 2 * 192 + 32;  // dwords per K-pair row; %64 == 32
  constexpr int NW = 8;

  __shared__ __attribute__((aligned(16))) float sB[(KPAD / 2) * PSTR];
  __shared__ float sBias[192];
  __shared__ __attribute__((aligned(16))) float sA[NW][16 * ASTR];

  const int tid = threadIdx.x;
  // Stage fused weights: cols 0..63 gate i, 64..127 gate c, 128..191 gate o.
  // Rows 0..D-1 = x part, D..D+3 = edge_attr part (skipping the zero h-rows).
  for (int t = tid; t < KPAD * 192; t += 256) {
    const int k = t / 192, col = t % 192;
    const int g = col >> 6, c = col & 63;
    const float* W = (g == 0) ? Wm0 : (g == 1) ? Wm1 : Wm2;
    float v = 0.0f;
    if (k < D) v = W[k * 64 + c];
    else if (k < D + 4) v = W[(D + 64 + (k - D)) * 64 + c];
    sB[(k >> 1) * PSTR + 2 * col + (k & 1)] = v;
  }
  if (tid < 192) {
    const int g = tid >> 6, c = tid & 63;
    const float* b = (g == 0) ? bm0 : (g == 1) ? bm1 : bm2;
    sBias[tid] = b[c];
  }
  __syncthreads();

  const int wave = tid >> 5, lane = tid & 31;
  const int half = lane >> 4, mn = lane & 15;
  float* A = &sA[wave][0];
  const int nTiles = (E + 15) >> 4;
  const int gstride = gridDim.x * NW;

  for (int tile = blockIdx.x * NW + wave; tile < nTiles; tile += gstride) {
    const int base = tile << 4;
    // Gather: lanes 0-15 fetch x[src] rows, lanes 16-31 fetch edge_attr + pad.
    if (half == 0) {
      const int e = base + mn;
      const int s = (e < E) ? src[e] : 0;
      const float* xr = x + (size_t)s * D;
      float* row = A + mn * ASTR;
      if constexpr (D % 4 == 0) {
#pragma unroll
        for (int j = 0; j < D; j += 4) {
          const float4 v4 = *reinterpret_cast<const float4*>(xr + j);
          row[j + 0] = v4.x; row[j + 1] = v4.y;
          row[j + 2] = v4.z; row[j + 3] = v4.w;
        }
      } else {
#pragma unroll
        for (int j = 0; j < D; ++j) row[j] = xr[j];
      }
    } else {
      const int e = base + mn;
      const float* er = eattr + 4 * (size_t)((e < E) ? e : 0);
      float* row = A + mn * ASTR;
      row[D + 0] = er[0]; row[D + 1] = er[1];
      row[D + 2] = er[2]; row[D + 3] = er[3];
#pragma unroll
      for (int j = K; j < KPAD; ++j) row[j] = 0.0f;
    }

    v8f acc[12] = {};
    for (int kk = 0; kk < KPAD; kk += 4) {
      const int k0 = kk + 2 * half;  // ISA A-layout: lanes>=16 take K pair 2,3
      const v2f a = *reinterpret_cast<const v2f*>(&A[mn * ASTR + k0]);
      const float* bp = &sB[(k0 >> 1) * PSTR + 2 * mn];
#pragma unroll
      for (int n = 0; n < 12; ++n) {
        const v2f b = *reinterpret_cast<const v2f*>(bp + 32 * n);
        acc[n] = wmma_f32(a, b, acc[n]);
      }
    }

    int dsts[8];
#pragma unroll
    for (int v = 0; v < 8; ++v) {
      const int e = base + v + 8 * half;  // C-layout: M = vgpr + 8*(lane/16)
      dsts[v] = (e < E) ? dst[e] : -1;
    }

#pragma unroll
    for (int n = 0; n < 12; ++n) {
      const int col = n * 16 + mn;
      const int g = col >> 6, hc = col & 63;
      const float bias = sBias[col];
      float* aggg = agg + (size_t)g * N * 64;
#pragma unroll
      for (int v = 0; v < 8; ++v) {
        const int d = dsts[v];
        if (d >= 0) {
          float m = acc[n][v] + bias;
          m = fmaxf(m, 0.0f);
          atomicAdd(aggg + (size_t)d * 64 + hc, m);
        }
      }
    }
  }
}

// ---------------------------------------------------------------------------
// Node kernel: fused 3-gate node update + LSTM activations.
//   out_g = [x, agg_g] @ Wn_g(effective) + bn_g   (h_prev rows skipped)
//   h = sigmoid(out_o) * tanh( sigmoid(out_i) * tanh(out_c) )   (c_prev == 0)
// Same K-pair-interleaved B layout (PSTR = 160 dwords, %64 == 32).
// ---------------------------------------------------------------------------
template <int D>
__launch_bounds__(128)
__global__ void node_update_kernel(const float* __restrict__ x,    // [N, D]
                                   const float* __restrict__ agg,  // [3][N*64]
                                   const float* __restrict__ Wn0,
                                   const float* __restrict__ bn0,
                                   const float* __restrict__ Wn1,
                                   const float* __restrict__ bn1,
                                   const float* __restrict__ Wn2,
                                   const float* __restrict__ bn2,
                                   float* __restrict__ hout,       // [N, 64]
                                   int N) {
  constexpr int DPAD = (D + 3) & ~3;
  constexpr int R = DPAD + 64;  // even
  constexpr int AXS = (DPAD % 16 == 0) ? DPAD + 4 : DPAD;
  constexpr int AAS = 68;
  constexpr int PSTR = 2 * 64 + 32;  // dwords per K-pair row; %64 == 32
  constexpr int NW = 4;

  __shared__ __attribute__((aligned(16))) float sB[3 * (R / 2) * PSTR];
  __shared__ float sBias[192];
  __shared__ __attribute__((aligned(16))) float sAx[NW][16 * AXS];
  __shared__ __attribute__((aligned(16))) float sAa[NW][16 * AAS];

  const int tid = threadIdx.x;
  for (int t = tid; t < 3 * R * 64; t += 128) {
    const int g = t / (R * 64);
    const int r = (t / 64) % R;
    const int c = t & 63;
    const float* W = (g == 0) ? Wn0 : (g == 1) ? Wn1 : Wn2;
    float v = 0.0f;
    if (r < D) v = W[r * 64 + c];                               // x part
    else if (r >= DPAD) v = W[(D + 64 + (r - DPAD)) * 64 + c];  // agg part
    sB[(g * (R / 2) + (r >> 1)) * PSTR + 2 * c + (r & 1)] = v;
  }
  for (int t = tid; t < 192; t += 128) {
    const int g = t >> 6, c = t & 63;
    const float* b = (g == 0) ? bn0 : (g == 1) ? bn1 : bn2;
    sBias[t] = b[c];
  }
  __syncthreads();

  const int wave = tid >> 5, lane = tid & 31;
  const int half = lane >> 4, mn = lane & 15;
  float* Ax = &sAx[wave][0];
  float* Aa = &sAa[wave][0];
  const int nTiles = (N + 15) >> 4;
  const int gstride = gridDim.x * NW;

  for (int tile = blockIdx.x * NW + wave; tile < nTiles; tile += gstride) {
    const int base = tile << 4;
    for (int idx = lane; idx < 16 * DPAD; idx += 32) {
      const int m = idx / DPAD, k = idx % DPAD;
      const int node = base + m;
      Ax[m * AXS + k] = (node < N && k < D) ? x[(size_t)node * D + k] : 0.0f;
    }

    v8f res[3][4];
    for (int g = 0; g < 3; ++g) {
      const float* ag = agg + (size_t)g * N * 64;
      for (int idx = lane; idx < 16 * 64; idx += 32) {
        const int m = idx >> 6, k = idx & 63;
        const int node = base + m;
        Aa[m * AAS + k] = (node < N) ? ag[(size_t)node * 64 + k] : 0.0f;
      }
      const float* Bg = sB + (size_t)g * (R / 2) * PSTR;
      v8f acc[4] = {};
      for (int kk = 0; kk < DPAD; kk += 4) {
        const int k0 = kk + 2 * half;
        const v2f a = *reinterpret_cast<const v2f*>(&Ax[mn * AXS + k0]);
        const float* bp = &Bg[(k0 >> 1) * PSTR + 2 * mn];
#pragma unroll
        for (int n = 0; n < 4; ++n) {
          const v2f b = *reinterpret_cast<const v2f*>(bp + 32 * n);
          acc[n] = wmma_f32(a, b, acc[n]);
        }
      }
      for (int kk = 0; kk < 64; kk += 4) {
        const int k0 = kk + 2 * half;
        const int r0 = DPAD + k0;  // even
        const v2f a = *reinterpret_cast<const v2f*>(&Aa[mn * AAS + k0]);
        const float* bp = &Bg[(r0 >> 1) * PSTR + 2 * mn];
#pragma unroll
        for (int n = 0; n < 4; ++n) {
          const v2f b = *reinterpret_cast<const v2f*>(bp + 32 * n);
          acc[n] = wmma_f32(a, b, acc[n]);
        }
      }
#pragma unroll
      for (int n = 0; n < 4; ++n) res[g][n] = acc[n];
    }

#pragma unroll
    for (int n = 0; n < 4; ++n) {
      const int col = n * 16 + mn;
#pragma unroll
      for (int v = 0; v < 8; ++v) {
        const int node = base + v + 8 * half;
        if (node < N) {
          const float gi = sigmoidf_(res[0][n][v] + sBias[col]);
          const float gc = tanhf(res[1][n][v] + sBias[64 + col]);
          const float go = sigmoidf_(res[2][n][v] + sBias[128 + col]);
          const float c = gi * gc;  // c_prev == 0 -> f gate is dead code
          hout[(size_t)node * 64 + col] = go * tanhf(c);
        }
      }
    }
  }
}

// ---------------------------------------------------------------------------
extern "C" void kernel_launch(void* const* d_in, const int* in_sizes, int n_in,
                              void* d_out, int out_size, void* d_ws,
                              size_t ws_size, hipStream_t stream) {
  const int N = in_sizes[0] / 26;
  const int E = in_sizes[1] / 2;
  const float* x0 = (const float*)d_in[0];
  const int* ei = (const int*)d_in[1];
  const int* src = ei;
  const int* dst = ei + E;
  const float* ea = (const float*)d_in[2];

  // params flattened: layer-major, gates in order i,f,c,o, each {Wm,bm,Wn,bn}
  auto P = [&](int layer, int gate, int which) -> const float* {
    return (const float*)d_in[3 + (layer * 4 + gate) * 4 + which];
  };
  // live gates only: i=0, c=2, o=3 (f is multiplied by c_prev == 0)

  float* agg = (float*)d_ws;                 // [3][N*64]
  float* hA = agg + (size_t)3 * N * 64;      // [N*64]
  float* hB = hA + (size_t)N * 64;           // [N*64]
  const size_t aggBytes = (size_t)3 * N * 64 * sizeof(float);

  const dim3 eb(256), eg(2048), nb(128), ng(512);

  // Layer 0 (D = 26)
  hipMemsetAsync(agg, 0, aggBytes, stream);
  edge_msg_kernel<26><<<eg, eb, 0, stream>>>(
      x0, src, dst, ea, P(0, 0, 0), P(0, 0, 1), P(0, 2, 0), P(0, 2, 1),
      P(0, 3, 0), P(0, 3, 1), agg, N, E);
  node_update_kernel<26><<<ng, nb, 0, stream>>>(
      x0, agg, P(0, 0, 2), P(0, 0, 3), P(0, 2, 2), P(0, 2, 3), P(0, 3, 2),
      P(0, 3, 3), hA, N);

  // Layer 1 (D = 64)
  hipMemsetAsync(agg, 0, aggBytes, stream);
  edge_msg_kernel<64><<<eg, eb, 0, stream>>>(
      hA, src, dst, ea, P(1, 0, 0), P(1, 0, 1), P(1, 2, 0), P(1, 2, 1),
      P(1, 3, 0), P(1, 3, 1), agg, N, E);
  node_update_kernel<64><<<ng, nb, 0, stream>>>(
      hA, agg, P(1, 0, 2), P(1, 0, 3), P(1, 2, 2), P(1, 2, 3), P(1, 3, 2),
      P(1, 3, 3), hB, N);

  // Layer 2 (D = 64) -> d_out
  hipMemsetAsync(agg, 0, aggBytes, stream);
  edge_msg_kernel<64><<<eg, eb, 0, stream>>>(
      hB, src, dst, ea, P(2, 0, 0), P(2, 0, 1), P(2, 2, 0), P(2, 2, 1),
      P(2, 3, 0), P(2, 3, 1), agg, N, E);
  node_update_kernel<64><<<ng, nb, 0, stream>>>(
      hB, agg, P(2, 0, 2), P(2, 0, 3), P(2, 2, 2), P(2, 2, 3), P(2, 3, 2),
      P(2, 3, 3), (float*)d_out, N);
}